// GCNEncoder_68685116997964
// MI455X (gfx1250) — compile-verified
//
#include <hip/hip_runtime.h>
#include <hip/hip_bf16.h>

typedef __attribute__((ext_vector_type(2))) float v2f;
typedef __attribute__((ext_vector_type(8))) float v8f;

// ---------------------------------------------------------------------------
// Degree count: deg[dst[e]] += 1  (f32 atomics into L2; N*4B = 400KB, L2 hit)
// ---------------------------------------------------------------------------
__global__ void deg_count_kernel(const long long* __restrict__ dst,
                                 float* __restrict__ deg, long long E) {
  long long i = (long long)blockIdx.x * blockDim.x + threadIdx.x;
  if (i < E) {
    long long d = dst[i];
    unsafeAtomicAdd(&deg[d], 1.0f);
  }
}

// dinv[i] = rsqrt(deg[i] + 1)   (+1 = self loop)
__global__ void deg_finalize_kernel(float* __restrict__ dinv, int N) {
  int i = blockIdx.x * blockDim.x + threadIdx.x;
  if (i < N) dinv[i] = rsqrtf(dinv[i] + 1.0f);
}

// Wcat[k][0:Nh] = Wa[k], Wcat[k][Nh:2Nh] = Wb[k]   (column-wise concat)
__global__ void pack_w_kernel(const float* __restrict__ Wa,
                              const float* __restrict__ Wb,
                              float* __restrict__ Wcat, int K, int Nh) {
  int i = blockIdx.x * blockDim.x + threadIdx.x;
  if (i >= K * Nh) return;
  int k = i / Nh, n = i % Nh;
  Wcat[(size_t)k * (2 * Nh) + n] = Wa[i];
  Wcat[(size_t)k * (2 * Nh) + Nh + n] = Wb[i];
}

// ---------------------------------------------------------------------------
// GEMM  D[M x NW] = X[M x K] @ W[K x NW]  via V_WMMA_F32_16X16X4_F32.
// LDS staging (K-chunked, KC=32):
//  - W chunk stored TRANSPOSED: ldsW[n * 36 + k]. A B fragment (rows k,k+1 of
//    column n) is then 2 contiguous floats -> single ds_load_b64 directly into
//    the WMMA operand pair (no v_mov marshalling), all tiles share one base
//    address + immediate offsets. Pitch 36: 36*n mod 64 permutes the (mod 4 ==
//    0) residues, so 16 column lanes hit 16 distinct bank pairs and the k+2
//    half-wave lands on disjoint (mod 4 == 2) banks -> conflict-free.
//  - Per-wave A strip [16 x KC] staged with 4 coalesced float4 loads, rows
//    padded to 36 floats, fragment read as one 8B ds load.
// Persistent blocks, grid-stride over 16-row strips; all WMMA branches are
// wave-uniform (EXEC all-ones); inactive waves still reach every barrier.
// Fragments: A 16x4 f32: lanes 0-15 row M=lane {K,K+1}, lanes 16-31 {K+2,K+3}.
//            B 4x16:     lanes 0-15 rows {K,K+1},      lanes 16-31 {K+2,K+3}.
//            C/D 16x16:  VGPR r -> M = r + 8*(lane>=16), N = lane&15.
// ---------------------------------------------------------------------------
template <int NTILES>
__global__ __launch_bounds__(256) void gemm_wmma_f32_kernel(
    const float* __restrict__ X, const float* __restrict__ W,
    float* __restrict__ D, int M, int K, int lda, int ldd) {
  constexpr int KC = 32;
  constexpr int NW = NTILES * 16;
  constexpr int KCP = KC + 4;  // transposed-W pitch (and A pitch): 36 floats
  __shared__ float ldsW[NW * KCP];
  __shared__ float ldsA[8 * 16 * KCP];

  const int lane = threadIdx.x & 31;
  const int waveIn = threadIdx.x >> 5;
  const int halfSel = lane >> 4;
  const int l16 = lane & 15;
  const int numStrips = M / 16;  // M divisible by 16 here

  for (int stripBase = blockIdx.x * 8; stripBase < numStrips;
       stripBase += gridDim.x * 8) {
    const int strip = stripBase + waveIn;
    const bool active = strip < numStrips;  // wave-uniform
    const int mBase = strip * 16;
    v8f acc[NTILES] = {};

    for (int kc = 0; kc < K; kc += KC) {
      __syncthreads();
      // stage W chunk transposed: coalesced float4 global read, b32 LDS scatter
      for (int idx = threadIdx.x * 4; idx < KC * NW; idx += 256 * 4) {
        int k = idx / NW, n = idx % NW;
        float4 wv = *(const float4*)(W + (size_t)(kc + k) * NW + n);
        ldsW[(n + 0) * KCP + k] = wv.x;
        ldsW[(n + 1) * KCP + k] = wv.y;
        ldsW[(n + 2) * KCP + k] = wv.z;
        ldsW[(n + 3) * KCP + k] = wv.w;
      }
      if (active) {
        // stage A strip: 16 rows x KC cols, 4 coalesced float4 loads
        float* la = &ldsA[waveIn * 16 * KCP];
#pragma unroll
        for (int i = 0; i < (16 * KC) / 128; ++i) {
          int idx = i * 128 + lane * 4;
          int row = idx / KC, col = idx % KC;
          float4 xv = *(const float4*)(X + (size_t)(mBase + row) * lda + kc + col);
          *(float4*)(&la[row * KCP + col]) = xv;
        }
      }
      __syncthreads();
      if (active) {
        const float* la = &ldsA[waveIn * 16 * KCP + l16 * KCP + 2 * halfSel];
        const float* lw = &ldsW[l16 * KCP + 2 * halfSel];
#pragma unroll
        for (int k2 = 0; k2 < KC; k2 += 4) {
          float2 av = *(const float2*)(la + k2);
          v2f a;
          a[0] = av.x;
          a[1] = av.y;
          v2f bf[NTILES];
#pragma unroll
          for (int t = 0; t < NTILES; ++t) {
            float2 bv = *(const float2*)(lw + t * 16 * KCP + k2);
            bf[t][0] = bv.x;
            bf[t][1] = bv.y;
          }
#pragma unroll
          for (int t = 0; t < NTILES; ++t) {
            acc[t] = __builtin_amdgcn_wmma_f32_16x16x4_f32(
                false, a, false, bf[t], (short)0, acc[t], false, false);
          }
        }
      }
    }

    if (active) {
#pragma unroll
      for (int t = 0; t < NTILES; ++t) {
        const int n = t * 16 + l16;
#pragma unroll
        for (int r = 0; r < 8; ++r) {
          const int m = mBase + r + 8 * halfSel;
          D[(size_t)m * ldd + n] = acc[t][r];
        }
      }
    }
  }
}

// ---------------------------------------------------------------------------
// Edge message pass: Agg[dst] += H[src] * dinv[src]*dinv[dst].
// One wave handles 32 edges: coalesced per-lane index/norm loads, then each
// edge is broadcast via __shfl while all 32 lanes move one float4 of the
// 512B row (coalesced gather + 4 f32 atomics/lane). H and Agg working sets
// are ~51MB each -> resident in the 192MB L2.
// ---------------------------------------------------------------------------
__global__ __launch_bounds__(256) void edge_scatter_kernel(
    const long long* __restrict__ src, const long long* __restrict__ dst,
    const float* __restrict__ dinv, const float* __restrict__ H, int ldh,
    float* __restrict__ Agg, long long E) {
  const int lane = threadIdx.x & 31;
  const long long wid = ((long long)blockIdx.x * blockDim.x + threadIdx.x) >> 5;
  const long long base = wid * 32;
  if (base >= E) return;
  const int cnt = (int)((E - base < 32) ? (E - base) : 32);
  long long s = 0, d = 0;
  float w = 0.0f;
  if (lane < cnt) {
    s = src[base + lane];
    d = dst[base + lane];
    w = dinv[s] * dinv[d];
  }
  for (int j = 0; j < cnt; ++j) {
    long long sj = __shfl(s, j);
    long long dj = __shfl(d, j);
    float wj = __shfl(w, j);
    float4 v = *(const float4*)(H + (size_t)sj * ldh + (size_t)lane * 4);
    float* out = Agg + (size_t)dj * 128 + (size_t)lane * 4;
    unsafeAtomicAdd(out + 0, v.x * wj);
    unsafeAtomicAdd(out + 1, v.y * wj);
    unsafeAtomicAdd(out + 2, v.z * wj);
    unsafeAtomicAdd(out + 3, v.w * wj);
  }
}

// hidden = relu(agg1 + h1*dinv^2 + b1) + hp + bp
// hAB rows are [h1(0:128) | hp(128:256)]; result overwrites cols 0:128.
__global__ __launch_bounds__(256) void combine_hidden_kernel(
    const float* __restrict__ agg1, const float* __restrict__ dinv,
    const float* __restrict__ b1, const float* __restrict__ bp,
    float* __restrict__ hAB, int N) {
  long long t = (long long)blockIdx.x * blockDim.x + threadIdx.x;
  long long i = t >> 5;
  int c4 = (int)(t & 31) * 4;
  if (i >= N) return;
  float di = dinv[i];
  float d2 = di * di;
  size_t row = (size_t)i * 256;
  float4 a = *(const float4*)(agg1 + (size_t)i * 128 + c4);
  float4 h = *(const float4*)(hAB + row + c4);
  float4 p = *(const float4*)(hAB + row + 128 + c4);
  float4 B1 = *(const float4*)(b1 + c4);
  float4 Bp = *(const float4*)(bp + c4);
  float4 r;
  r.x = fmaxf(fmaf(h.x, d2, a.x) + B1.x, 0.0f) + p.x + Bp.x;
  r.y = fmaxf(fmaf(h.y, d2, a.y) + B1.y, 0.0f) + p.y + Bp.y;
  r.z = fmaxf(fmaf(h.z, d2, a.z) + B1.z, 0.0f) + p.z + Bp.z;
  r.w = fmaxf(fmaf(h.w, d2, a.w) + B1.w, 0.0f) + p.w + Bp.w;
  *(float4*)(hAB + row + c4) = r;
}

// out = agg + hml*dinv^2 + bias; cols 0..63 -> mu, 64..127 -> logstd.
// hml has row stride 256 (packed hmu|hls region of bufAB).
__global__ __launch_bounds__(256) void finalize_kernel(
    const float* __restrict__ agg, const float* __restrict__ hml,
    const float* __restrict__ dinv, const float* __restrict__ bmu,
    const float* __restrict__ bls, float* __restrict__ out, int N) {
  long long t = (long long)blockIdx.x * blockDim.x + threadIdx.x;
  long long i = t >> 5;
  int c4 = (int)(t & 31) * 4;
  if (i >= N) return;
  float di = dinv[i];
  float d2 = di * di;
  float4 a = *(const float4*)(agg + (size_t)i * 128 + c4);
  float4 h = *(const float4*)(hml + (size_t)i * 256 + c4);
  float4 bias;
  float* dstp;
  if (c4 < 64) {
    bias = *(const float4*)(bmu + c4);
    dstp = out + (size_t)i * 64 + c4;
  } else {
    bias = *(const float4*)(bls + (c4 - 64));
    dstp = out + (size_t)N * 64 + (size_t)i * 64 + (c4 - 64);
  }
  float4 r;
  r.x = fmaf(h.x, d2, a.x) + bias.x;
  r.y = fmaf(h.y, d2, a.y) + bias.y;
  r.z = fmaf(h.z, d2, a.z) + bias.z;
  r.w = fmaf(h.w, d2, a.w) + bias.w;
  *(float4*)dstp = r;
}

// ---------------------------------------------------------------------------
extern "C" void kernel_launch(void* const* d_in, const int* in_sizes, int n_in,
                              void* d_out, int out_size, void* d_ws, size_t ws_size,
                              hipStream_t stream) {
  const float* x   = (const float*)d_in[0];
  const long long* eidx = (const long long*)d_in[1];  // int64 [2,E]
  const float* W1  = (const float*)d_in[2];
  const float* b1  = (const float*)d_in[3];
  const float* Wp  = (const float*)d_in[4];
  const float* bp  = (const float*)d_in[5];
  const float* Wmu = (const float*)d_in[6];
  const float* bmu = (const float*)d_in[7];
  const float* Wls = (const float*)d_in[8];
  const float* bls = (const float*)d_in[9];

  const int HID = in_sizes[3];        // 128
  const int OUT = in_sizes[7];        // 64
  const int IN  = in_sizes[2] / HID;  // 256
  const int N   = in_sizes[0] / IN;   // 100000 (divisible by 16)
  const long long E = (long long)in_sizes[1] / 2;

  const long long* src = eidx;
  const long long* dst = eidx + E;

  // workspace: dinv[Npad] | bufAB[N,256] | bufC[N,128] | Wcat1[256,256] |
  //            Wcat2[128,128]                                     (~154 MB)
  float* ws = (float*)d_ws;
  size_t Npad = ((size_t)N + 127) & ~(size_t)127;
  float* dinv  = ws;
  float* bufAB = ws + Npad;                  // h1|hp, then hidden|hmu|hls
  float* bufC  = bufAB + (size_t)N * 256;    // agg1, later agg(mu|ls)
  float* Wcat1 = bufC + (size_t)N * 128;     // [IN, 2*HID]
  float* Wcat2 = Wcat1 + (size_t)IN * 2 * HID;  // [HID, 2*OUT]

  hipMemsetAsync(dinv, 0, (size_t)N * sizeof(float), stream);
  hipMemsetAsync(bufC, 0, (size_t)N * 128 * sizeof(float), stream);

  // pack W1|Wp -> Wcat1, Wmu|Wls -> Wcat2
  pack_w_kernel<<<(IN * HID + 255) / 256, 256, 0, stream>>>(W1, Wp, Wcat1, IN, HID);
  pack_w_kernel<<<(HID * OUT + 255) / 256, 256, 0, stream>>>(Wmu, Wls, Wcat2, HID, OUT);

  // degrees -> dinv
  {
    long long bl = (E + 255) / 256;
    deg_count_kernel<<<(int)bl, 256, 0, stream>>>(dst, dinv, E);
    deg_finalize_kernel<<<(N + 255) / 256, 256, 0, stream>>>(dinv, N);
  }

  const int numStrips = N / 16;
  int gemmBlocks = (numStrips + 7) / 8;
  if (gemmBlocks > 720) gemmBlocks = 720;  // persistent blocks, grid-stride

  // [h1 | hp] = x @ Wcat1 -> bufAB  (single pass over x, 16x256 per wave)
  gemm_wmma_f32_kernel<16><<<gemmBlocks, 256, 0, stream>>>(
      x, Wcat1, bufAB, N, IN, IN, 256);

  // agg1 = A_norm @ h1 (edge scatter; h1 = bufAB cols 0:128, stride 256)
  {
    long long waves = (E + 31) / 32;
    long long bl = (waves * 32 + 255) / 256;
    edge_scatter_kernel<<<(int)bl, 256, 0, stream>>>(src, dst, dinv, bufAB, 256, bufC, E);
  }

  // hidden = relu(agg1 + h1*dinv2 + b1) + hp + bp  (into bufAB cols 0:128)
  {
    long long thr = (long long)N * 32;
    combine_hidden_kernel<<<(int)((thr + 255) / 256), 256, 0, stream>>>(
        bufC, dinv, b1, bp, bufAB, N);
  }

  // [hmu | hls] = hidden @ Wcat2 -> bufAB cols 128:256 (read cols 0:128,
  // write cols 128:256 of the same rows -> disjoint, safe)
  gemm_wmma_f32_kernel<8><<<gemmBlocks, 256, 0, stream>>>(
      bufAB, Wcat2, bufAB + 128, N, HID, 256, 256);

  // fused edge pass for both mu and logstd
  hipMemsetAsync(bufC, 0, (size_t)N * 128 * sizeof(float), stream);
  {
    long long waves = (E + 31) / 32;
    long long bl = (waves * 32 + 255) / 256;
    edge_scatter_kernel<<<(int)bl, 256, 0, stream>>>(src, dst, dinv, bufAB + 128, 256, bufC, E);
  }

  // out = (mu [N,64], logstd [N,64]) flat
  {
    long long thr = (long long)N * 32;
    finalize_kernel<<<(int)((thr + 255) / 256), 256, 0, stream>>>(
        bufC, bufAB + 128, dinv, bmu, bls, (float*)d_out, N);
  }
}